// SingleNode_Reserve_DRO_DA_OptNet_46926812676867
// MI455X (gfx1250) — compile-verified
//
#include <hip/hip_runtime.h>

// ---------------------------------------------------------------------------
// SingleNode Reserve DRO solver, fully fused for gfx1250 (MI455X).
//  - grid = B(=8) blocks, 256 threads (8 wave32 waves) each; one block solves
//    one batch element end-to-end (40 PGD iterations + final evaluation).
//  - small state lives in LDS (~141 KB of the 320 KB/WGP pool).
//  - d_up / d_dn ([T][G][N] f32, n fastest) live in d_ws (L2-resident).
//  - G-axis reduction einsums run on V_WMMA_F32_16X16X4_F32 with a two-column
//    weight matrix (col0 = +/-1 -> r2 residual, col1 = rtu/rtd -> Qn einsum).
// ---------------------------------------------------------------------------

typedef __attribute__((ext_vector_type(2))) float v2f;
typedef __attribute__((ext_vector_type(8))) float v8f;

#define B_    8
#define N_    128
#define G_    100
#define T_    24
#define NIT   40
#define GT    (G_ * T_)        // 2400
#define NT    (N_ * T_)        // 3072
#define GN    (G_ * N_)        // 12800
#define TGN   (T_ * G_ * N_)   // 307200

#define LRATE 2.0e-4f
#define RHO_  10.0f
#define VOLL_ 1000.0f
#define VOSP_ 50.0f

// d_out layout (floats): P | R_up | R_dn | Cost | phi | gamma | obj
#define OFF_P    0
#define OFF_RUP  (B_ * GT)           // 19200
#define OFF_RDN  (2 * B_ * GT)       // 38400
#define OFF_COST (3 * B_ * GT)       // 57600
#define OFF_PHI  (4 * B_ * GT)       // 76800
#define OFF_GAM  (OFF_PHI + B_ * N_) // 77824
#define OFF_OBJ  (OFF_GAM + B_)      // 77832

static __device__ inline v8f wmma_k4(v2f a, v2f b, v8f c) {
  // V_WMMA_F32_16X16X4_F32 : D(16x16,f32) = A(16x4,f32) x B(4x16,f32) + C
  return __builtin_amdgcn_wmma_f32_16x16x4_f32(
      /*neg_a=*/false, a, /*neg_b=*/false, b,
      /*c_mod=*/(short)0, c, /*reuse_a=*/false, /*reuse_b=*/false);
}

static __device__ inline float blockReduceSum(float v, float* sRed, int tid) {
  sRed[tid] = v;
  __syncthreads();
#pragma unroll
  for (int s = 128; s > 0; s >>= 1) {
    if (tid < s) sRed[tid] += sRed[tid + s];
    __syncthreads();
  }
  float r = sRed[0];
  __syncthreads();
  return r;
}

// jax maximum() subgradient weight on x: x>y -> 1, x<y -> 0, tie -> 0.5
static __device__ inline float maxw(float x, float y) {
  return (x > y) ? 1.0f : ((x < y) ? 0.0f : 0.5f);
}
static __device__ inline float sgnf(float x) {
  return (x > 0.0f) ? 1.0f : ((x < 0.0f) ? -1.0f : 0.0f);
}

__global__ __launch_bounds__(256) void dro_solver_kernel(
    const float* __restrict__ forecast,  // [B,T]
    const float* __restrict__ omega,     // [B,N,T]
    const float* __restrict__ om_min,    // [B,T]
    const float* __restrict__ om_max,    // [B,T]
    const float* __restrict__ epsv,      // [B]
    const float* __restrict__ pminG,     // [G]
    const float* __restrict__ pmaxG,     // [G]
    const float* __restrict__ bG,        // [G]
    const float* __restrict__ cG,        // [G]
    float* __restrict__ out,
    float* __restrict__ ws) {
  const int tid  = threadIdx.x;
  const int b    = blockIdx.x;
  const int lane = tid & 31;
  const int wid  = tid >> 5;

  // ---- LDS state ----------------------------------------------------------
  __shared__ __align__(16) float sP[GT], sRup[GT], sRdn[GT];
  __shared__ __align__(16) float sUmx[GT], sDmx[GT], sUmn[GT], sDmn[GT];
  __shared__ __align__(16) float sLS[NT], sSPa[NT], sOmg[NT], sR2[NT];
  __shared__ __align__(16) float sGRu[GT], sGRd[GT];
  __shared__ __align__(16) float sQn[N_], sW1[N_], sW2[N_], sW3[N_];
  __shared__ __align__(16) float sDistMx[N_], sDistMn[N_];
  __shared__ __align__(16) float sR1[T_], sR3[T_], sR4[T_];
  __shared__ __align__(16) float sFc[T_], sOmx[T_], sOmn[T_];
  __shared__ __align__(16) float sLSm[T_], sSPm[T_], sLSn[T_], sSPn[T_];
  __shared__ __align__(16) float sPmn[G_], sPmx[G_], sBB[G_], sCC[G_];
  __shared__ __align__(16) float sRtu[G_], sRtd[G_], sRupP[G_], sRdnP[G_], sRamp[G_];
  __shared__ __align__(16) float sRed[256];
  __shared__ float sGamma[1];

  float* DUP = ws + (size_t)b * TGN;                     // [T][G][N]
  float* DDN = ws + (size_t)B_ * TGN + (size_t)b * TGN;  // [T][G][N]

  const float epsb  = epsv[b];
  const float invN  = 1.0f / (float)N_;
  const float tworo = 2.0f * RHO_;

  // ---- init ---------------------------------------------------------------
  for (int g = tid; g < G_; g += 256) {
    const float bb = bG[g];
    sPmn[g]  = pminG[g];
    sPmx[g]  = pmaxG[g];
    sBB[g]   = bb;
    sCC[g]   = cG[g];
    sRtu[g]  = 2.0f * bb;   // RT_UP * b
    sRtd[g]  = 0.5f * bb;   // RT_DN * b
    sRupP[g] = 0.05f * bb;  // RES_UP * b
    sRdnP[g] = 0.02f * bb;  // RES_DN * b
    sRamp[g] = pmaxG[g];    // RAMP_F * pmax
  }
  for (int t = tid; t < T_; t += 256) {
    sFc[t]  = forecast[b * T_ + t];
    sOmx[t] = om_max[b * T_ + t];
    sOmn[t] = om_min[b * T_ + t];
    sLSm[t] = 0.0f; sSPm[t] = 0.0f; sLSn[t] = 0.0f; sSPn[t] = 0.0f;
  }
  for (int c = tid; c < NT; c += 256) {
    sOmg[c] = omega[(size_t)b * NT + c];
    sLS[c] = 0.0f; sSPa[c] = 0.0f;
  }
  for (int c = tid; c < GT; c += 256) {
    const int g = c / T_;
    sP[c]   = 0.5f * (pminG[g] + pmaxG[g]);
    sRup[c] = 0.0f; sRdn[c] = 0.0f;
    sUmx[c] = 0.0f; sDmx[c] = 0.0f; sUmn[c] = 0.0f; sDmn[c] = 0.0f;
  }
  {
    const float4 z4 = make_float4(0.f, 0.f, 0.f, 0.f);
    float4* up4 = (float4*)DUP;
    float4* dn4 = (float4*)DDN;
    for (int e = tid; e < TGN / 4; e += 256) { up4[e] = z4; dn4[e] = z4; }
  }
  if (tid == 0) sGamma[0] = 0.0f;
  __syncthreads();

  if (tid < N_) {
    float dmx = 0.0f, dmn = 0.0f;
#pragma unroll
    for (int t = 0; t < T_; ++t) {
      const float o = sOmg[tid * T_ + t];
      dmx += sOmx[t] - o;
      dmn += o - sOmn[t];
    }
    sDistMx[tid] = dmx;
    sDistMn[tid] = dmn;
  }
  __syncthreads();

  float Qmax = 0.0f, Qmin = 0.0f;

  // WMMA lane-constant fragment selectors (column 0 -> sums, column 1 -> einsum)
  const int mrow = lane & 15;   // A row / B column
  const int khalf = lane >> 4;  // K half-select per ISA 16x4 / 4x16 layouts
  const float sel0 = (mrow == 0) ? 1.0f : 0.0f;
  const float sel1 = (mrow == 1) ? 1.0f : 0.0f;

  // ---- PGD iterations (iteration NIT only evaluates, then breaks) ---------
  for (int it = 0; it <= NIT; ++it) {
    if (tid < N_) sQn[tid] = 0.0f;
    __syncthreads();

    // Phase 1: WMMA reductions over g.
    //   col 0 of C: sum_g (d_up - d_dn)  -> r2[n,t]
    //   col 1 of C: sum_g rtu*d_up + rtd*d_dn -> Qn accumulation
    {
      const int n0 = wid * 16;
      const int laneOff = n0 + mrow;
      for (int t = 0; t < T_; ++t) {
        v8f acc = {};
        const float* baseU = DUP + (size_t)t * GN;
        const float* baseD = DDN + (size_t)t * GN;
        int idx = (2 * khalf) * N_ + laneOff;   // advances by 4*N per chunk
        for (int c4 = 0; c4 < G_; c4 += 4, idx += 4 * N_) {
          const int ga = c4 + 2 * khalf;
          v2f au, ad, bu, bd;
          au.x = baseU[idx];
          au.y = baseU[idx + N_];
          ad.x = baseD[idx];
          ad.y = baseD[idx + N_];
          // branch-free B fragment: col0 = +/-1, col1 = rtu/rtd, rest 0
          const v2f wu = *(const v2f*)(sRtu + ga);
          const v2f wd = *(const v2f*)(sRtd + ga);
          bu.x = sel0 + sel1 * wu.x;
          bu.y = sel0 + sel1 * wu.y;
          bd.x = sel1 * wd.x - sel0;
          bd.y = sel1 * wd.y - sel0;
          acc = wmma_k4(au, bu, acc);
          acc = wmma_k4(ad, bd, acc);
        }
        // C[m][j] lives in lane (j + 16*(m>=8)), VGPR (m&7).
        if (lane == 0 || lane == 16) {           // j = 0 : r2 residual
          const int nb = n0 + ((lane == 16) ? 8 : 0);
#pragma unroll
          for (int r = 0; r < 8; ++r) {
            const int ci = (nb + r) * T_ + t;
            sR2[ci] = acc[r] + sLS[ci] - sSPa[ci] - sOmg[ci];
          }
        }
        if (lane == 1 || lane == 17) {           // j = 1 : Qn part
          const int nb = n0 + ((lane == 17) ? 8 : 0);
#pragma unroll
          for (int r = 0; r < 8; ++r) {
            const int n  = nb + r;
            const int ci = n * T_ + t;
            sQn[n] += acc[r] + VOLL_ * sLS[ci] + VOSP_ * sSPa[ci];
          }
        }
      }
    }
    __syncthreads();

    // Phase 1b: Qmax / Qmin scalars.
    {
      float pmx = 0.0f, pmn = 0.0f;
      for (int c = tid; c < GT; c += 256) {
        const int g = c / T_;
        pmx += sRtu[g] * sUmx[c] + sRtd[g] * sDmx[c];
        pmn += sRtu[g] * sUmn[c] + sRtd[g] * sDmn[c];
      }
      if (tid < T_) {
        pmx += VOLL_ * sLSm[tid] + VOSP_ * sSPm[tid];
        pmn += VOLL_ * sLSn[tid] + VOSP_ * sSPn[tid];
      }
      Qmax = blockReduceSum(pmx, sRed, tid);
      Qmin = blockReduceSum(pmn, sRed, tid);
    }
    if (it == NIT) break;  // final state: sQn/Qmax/Qmin now hold v_final

    // Epigraph branch subgradient weights (jnp.maximum tie = 0.5).
    const float gamOld = sGamma[0];
    if (tid < N_) {
      const float v1 = sQn[tid];
      const float v2 = Qmax - gamOld * sDistMx[tid];
      const float v3 = Qmin - gamOld * sDistMn[tid];
      const float s23 = maxw(v2, v3);
      const float m23 = fmaxf(v2, v3);
      const float a   = maxw(v1, m23);
      sW1[tid] = a;
      sW2[tid] = (1.0f - a) * s23;
      sW3[tid] = (1.0f - a) * (1.0f - s23);
    }
    __syncthreads();
    const float S2 = blockReduceSum((tid < N_) ? sW2[tid] : 0.0f, sRed, tid);
    const float S3 = blockReduceSum((tid < N_) ? sW3[tid] : 0.0f, sRed, tid);
    const float Sg = blockReduceSum(
        (tid < N_) ? (sW2[tid] * sDistMx[tid] + sW3[tid] * sDistMn[tid]) : 0.0f,
        sRed, tid);

    // Phase 2: elementwise d_up/d_dn update + sum_n relu(d-R).
    // Wave owns one generator g per step; lane owns 4 consecutive scenarios,
    // so global traffic is b128 and the n-reduction finishes inside the wave
    // (single non-atomic LDS store per (g,t), no zero-init pass needed).
    for (int t = 0; t < T_; ++t) {
      const size_t slab = (size_t)t * GN;
      for (int gg = wid; gg < G_; gg += 8) {
        const size_t e = slab + (size_t)gg * N_ + (size_t)(lane * 4);
        if (gg + 8 < G_) {
          __builtin_prefetch(DUP + e + 8 * N_, 0, 0);
          __builtin_prefetch(DDN + e + 8 * N_, 0, 0);
        }
        const int gt = gg * T_ + t;
        float4 du4 = *(const float4*)(DUP + e);
        float4 dd4 = *(const float4*)(DDN + e);
        const float ru = sRup[gt], rd = sRdn[gt];
        const float rtug = sRtu[gg], rtdg = sRtd[gg];
        float du_[4] = {du4.x, du4.y, du4.z, du4.w};
        float dd_[4] = {dd4.x, dd4.y, dd4.z, dd4.w};
        float su = 0.0f, sd = 0.0f;
#pragma unroll
        for (int j = 0; j < 4; ++j) {
          const int n = lane * 4 + j;
          const float du = du_[j], dd = dd_[j];
          const float relu_u = fmaxf(du - ru, 0.0f);
          const float relu_d = fmaxf(dd - rd, 0.0f);
          const float r2v = sR2[n * T_ + t];
          const float w1n = sW1[n] * invN;
          const float gdu = w1n * rtug + tworo * (r2v + relu_u);
          const float gdd = w1n * rtdg + tworo * (-r2v + relu_d);
          du_[j] = fmaxf(du - LRATE * gdu, 0.0f);
          dd_[j] = fmaxf(dd - LRATE * gdd, 0.0f);
          su += relu_u;
          sd += relu_d;
        }
        *(float4*)(DUP + e) = make_float4(du_[0], du_[1], du_[2], du_[3]);
        *(float4*)(DDN + e) = make_float4(dd_[0], dd_[1], dd_[2], dd_[3]);
#pragma unroll
        for (int off = 16; off > 0; off >>= 1) {
          su += __shfl_down(su, off, 32);
          sd += __shfl_down(sd, off, 32);
        }
        if (lane == 0) { sGRu[gt] = su; sGRd[gt] = sd; }
      }
    }
    __syncthreads();

    // Phase 3a: per-t equality residuals.
    if (tid < T_) {
      const int t = tid;
      float s1 = 0.0f, s3 = 0.0f, s4 = 0.0f;
      for (int g = 0; g < G_; ++g) {
        const int c = g * T_ + t;
        s1 += sP[c];
        s3 += sUmx[c] - sDmx[c];
        s4 += sUmn[c] - sDmn[c];
      }
      sR1[t] = s1 - sFc[t];
      sR3[t] = s3 + sLSm[t] - sSPm[t] - sOmx[t];
      sR4[t] = s4 + sLSn[t] - sSPn[t] - sOmn[t];
    }
    __syncthreads();

    // Phase 3b: small-tensor updates (grads at OLD values, write after barrier)
    float nP[10], nRu[10], nRd[10], nUmx[10], nDmx[10], nUmn[10], nDmn[10];
#pragma unroll
    for (int k = 0; k < 10; ++k) {
      const int c = tid + 256 * k;
      if (c < GT) {
        const int g = c / T_;
        const int t = c - g * T_;
        const float Pc = sP[c], Ru = sRup[c], Rd = sRdn[c];
        const float reluUp = fmaxf(Pc + Ru - sPmx[g], 0.0f);
        const float reluDn = fmaxf(sPmn[g] - Pc + Rd, 0.0f);
        float gramp = 0.0f;
        if (t < T_ - 1) {
          const float d = sP[c + 1] - Pc;
          gramp -= fmaxf(fabsf(d) - sRamp[g], 0.0f) * sgnf(d);
        }
        if (t > 0) {
          const float d = Pc - sP[c - 1];
          gramp += fmaxf(fabsf(d) - sRamp[g], 0.0f) * sgnf(d);
        }
        const float gP = sBB[g] + tworo * (sR1[t] + reluUp - reluDn + gramp);
        nP[k] = fminf(fmaxf(Pc - LRATE * gP, sPmn[g]), sPmx[g]);

        const float umx = sUmx[c], dmx = sDmx[c], umn = sUmn[c], dmn = sDmn[c];
        const float rUmx = fmaxf(umx - Ru, 0.0f), rUmn = fmaxf(umn - Ru, 0.0f);
        const float rDmx = fmaxf(dmx - Rd, 0.0f), rDmn = fmaxf(dmn - Rd, 0.0f);

        const float gRu = sRupP[g] + tworo * (reluUp - (sGRu[c] + rUmx + rUmn));
        nRu[k] = fmaxf(Ru - LRATE * gRu, 0.0f);
        const float gRd = sRdnP[g] + tworo * (reluDn - (sGRd[c] + rDmx + rDmn));
        nRd[k] = fmaxf(Rd - LRATE * gRd, 0.0f);

        nUmx[k] = fmaxf(umx - LRATE * (S2 * invN * sRtu[g] + tworo * (sR3[t] + rUmx)), 0.0f);
        nDmx[k] = fmaxf(dmx - LRATE * (S2 * invN * sRtd[g] + tworo * (-sR3[t] + rDmx)), 0.0f);
        nUmn[k] = fmaxf(umn - LRATE * (S3 * invN * sRtu[g] + tworo * (sR4[t] + rUmn)), 0.0f);
        nDmn[k] = fmaxf(dmn - LRATE * (S3 * invN * sRtd[g] + tworo * (-sR4[t] + rDmn)), 0.0f);
      }
    }
    float nLS[12], nSP[12];
#pragma unroll
    for (int k = 0; k < 12; ++k) {
      const int c = tid + 256 * k;
      if (c < NT) {
        const int n = c / T_;
        const float w1n = sW1[n] * invN;
        const float r2v = sR2[c];
        nLS[k] = fmaxf(sLS[c]  - LRATE * (w1n * VOLL_ + tworo * r2v), 0.0f);
        nSP[k] = fmaxf(sSPa[c] - LRATE * (w1n * VOSP_ - tworo * r2v), 0.0f);
      }
    }
    float nLSm = 0.0f, nSPm = 0.0f, nLSn = 0.0f, nSPn = 0.0f;
    if (tid < T_) {
      const int t = tid;
      nLSm = fmaxf(sLSm[t] - LRATE * (S2 * invN * VOLL_ + tworo * sR3[t]), 0.0f);
      nSPm = fmaxf(sSPm[t] - LRATE * (S2 * invN * VOSP_ - tworo * sR3[t]), 0.0f);
      nLSn = fmaxf(sLSn[t] - LRATE * (S3 * invN * VOLL_ + tworo * sR4[t]), 0.0f);
      nSPn = fmaxf(sSPn[t] - LRATE * (S3 * invN * VOSP_ - tworo * sR4[t]), 0.0f);
    }
    float nGam = 0.0f;
    if (tid == 0) {
      const float gG = epsb - Sg * invN;
      nGam = fmaxf(sGamma[0] - LRATE * gG, 0.0f);
    }
    __syncthreads();
#pragma unroll
    for (int k = 0; k < 10; ++k) {
      const int c = tid + 256 * k;
      if (c < GT) {
        sP[c] = nP[k];   sRup[c] = nRu[k]; sRdn[c] = nRd[k];
        sUmx[c] = nUmx[k]; sDmx[c] = nDmx[k];
        sUmn[c] = nUmn[k]; sDmn[c] = nDmn[k];
      }
    }
#pragma unroll
    for (int k = 0; k < 12; ++k) {
      const int c = tid + 256 * k;
      if (c < NT) { sLS[c] = nLS[k]; sSPa[c] = nSP[k]; }
    }
    if (tid < T_) { sLSm[tid] = nLSm; sSPm[tid] = nSPm; sLSn[tid] = nLSn; sSPn[tid] = nSPn; }
    if (tid == 0) sGamma[0] = nGam;
    __syncthreads();
  }

  // ---- finalize: phi, obj, outputs (state is final; Qn/Qmax/Qmin are final)
  const float gam = sGamma[0];
  float phiPart = 0.0f;
  if (tid < N_) {
    const float v1 = sQn[tid];
    const float v2 = Qmax - gam * sDistMx[tid];
    const float v3 = Qmin - gam * sDistMn[tid];
    const float ph = fmaxf(v1, fmaxf(v2, v3));
    out[OFF_PHI + b * N_ + tid] = ph;
    phiPart = ph;
  }
  const float phiSum = blockReduceSum(phiPart, sRed, tid);

  float s1p = 0.0f;
  for (int c = tid; c < GT; c += 256) {
    const int g = c / T_;
    s1p += sBB[g] * sP[c] + sCC[g] + sRupP[g] * sRup[c] + sRdnP[g] * sRdn[c];
  }
  const float stage1 = blockReduceSum(s1p, sRed, tid);
  if (tid == 0) {
    out[OFF_GAM + b] = gam;
    out[OFF_OBJ + b] = stage1 + phiSum * invN + epsb * gam;
  }
  for (int c = tid; c < GT; c += 256) {
    const int g = c / T_;
    out[OFF_P    + b * GT + c] = sP[c];
    out[OFF_RUP  + b * GT + c] = sRup[c];
    out[OFF_RDN  + b * GT + c] = sRdn[c];
    out[OFF_COST + b * GT + c] = sBB[g] * sP[c] + sCC[g];
  }
}

extern "C" void kernel_launch(void* const* d_in, const int* in_sizes, int n_in,
                              void* d_out, int out_size, void* d_ws, size_t ws_size,
                              hipStream_t stream) {
  const float* forecast = (const float*)d_in[0];
  const float* omega    = (const float*)d_in[1];
  const float* om_min   = (const float*)d_in[2];
  const float* om_max   = (const float*)d_in[3];
  const float* epsv     = (const float*)d_in[4];
  const float* pmin     = (const float*)d_in[5];
  const float* pmax     = (const float*)d_in[6];
  const float* bG       = (const float*)d_in[7];
  const float* cG       = (const float*)d_in[8];
  float* out = (float*)d_out;
  float* ws  = (float*)d_ws;  // needs 2 * B * T * G * N floats (~19.7 MB)

  dro_solver_kernel<<<dim3(B_), dim3(256), 0, stream>>>(
      forecast, omega, om_min, om_max, epsv, pmin, pmax, bG, cG, out, ws);
}